// TradeArbitrageLayer_15968688406657
// MI455X (gfx1250) — compile-verified
//
#include <hip/hip_runtime.h>
#include <hip/hip_bf16.h>

#define N_NODES 1024
#define DIM 128
#define W1_LD 130   // W1 is (D, D+2) row-major

typedef __attribute__((ext_vector_type(2))) float v2f;
typedef __attribute__((ext_vector_type(8))) float v8f;

// ---------------------------------------------------------------------------
// C[m,n] = sum_k A[m*lda+k] * B[n*ldb+k]  (+ bias[n] * (rowscale ? rowscale[m] : 1))
// Tiled with V_WMMA_F32_16X16X4_F32 (full f32 precision, matches reference).
// One wave per 16x16 C tile; wave32 lane mapping per CDNA5 ISA 7.12.2:
//   A 16x4 : lane L holds (M=L%16, K = (L>=16?2:0)+{0,1}) in 2 VGPRs
//   B 4x16 : lane L holds (N=L%16, K = (L>=16?2:0)+{0,1}) in 2 VGPRs
//   C 16x16: VGPR v, lane L -> (M = v + (L>=16?8:0), N = L%16)
// ---------------------------------------------------------------------------
__global__ void __launch_bounds__(256)
wmma_gemm_nt(const float* __restrict__ A, int lda,
             const float* __restrict__ B, int ldb,
             float* __restrict__ C, int ldc,
             const float* __restrict__ bias,
             const float* __restrict__ rowscale,
             int K, int tilesN)
{
    const int lane = threadIdx.x & 31;
    const int wave = threadIdx.x >> 5;
    const int wid  = blockIdx.x * 8 + wave;
    const int tM = wid / tilesN;
    const int tN = wid % tilesN;

    const int halfSel = lane >> 4;   // 0: K0/K1, 1: K2/K3
    const int r       = lane & 15;
    const int m0 = tM * 16;
    const int n0 = tN * 16;

    const float* __restrict__ aRow = A + (size_t)(m0 + r) * lda + halfSel * 2;
    const float* __restrict__ bRow = B + (size_t)(n0 + r) * ldb + halfSel * 2;

    v8f c = {};
    for (int k = 0; k < K; k += 4) {
        v2f av = *(const v2f*)(aRow + k);
        v2f bv = *(const v2f*)(bRow + k);
        c = __builtin_amdgcn_wmma_f32_16x16x4_f32(
                /*neg_a=*/false, av, /*neg_b=*/false, bv,
                /*c_mod=*/(short)0, c, /*reuse_a=*/false, /*reuse_b=*/false);
    }

    const int colN = n0 + r;
    const float biasV = bias ? bias[colN] : 0.0f;
#pragma unroll
    for (int v = 0; v < 8; ++v) {
        const int row = m0 + v + halfSel * 8;
        const float rs = rowscale ? rowscale[row] : 1.0f;
        C[(size_t)row * ldc + colN] = c[v] + biasV * rs;
    }
}

// ---------------------------------------------------------------------------
// s1[m] = dot(Wh[m,:], a[0:128]);  s2[m] = dot(Wh[m,:], a[128:256])
// One block per row; 128 threads per half.
// ---------------------------------------------------------------------------
__global__ void __launch_bounds__(256)
s1s2_kernel(const float* __restrict__ Wh, const float* __restrict__ a,
            float* __restrict__ s1, float* __restrict__ s2)
{
    __shared__ float red[256];
    const int m    = blockIdx.x;
    const int t    = threadIdx.x & 127;
    const int half = threadIdx.x >> 7;

    red[threadIdx.x] = Wh[(size_t)m * DIM + t] * a[half * DIM + t];
    __syncthreads();
    for (int s = 64; s > 0; s >>= 1) {
        if (t < s) red[threadIdx.x] += red[threadIdx.x + s];
        __syncthreads();
    }
    if (t == 0) (half == 0 ? s1 : s2)[m] = red[threadIdx.x];
}

// ---------------------------------------------------------------------------
// Per row i:
//   e[j]   = leaky_relu(s1[i]+s2[j], 0.2) where adj>0 else -inf (exp underflows to 0)
//   p[j]   = exp(e[j]-max)  (exactly 0 for masked entries, matching reference)
//   acc[d] = (1/denom) * sum_j p[j]*relu(base[j,d] + pd[i,j]*w_pd[d] + ut[i,j]*w_ut[d])
//   rowsum[i] = 1 if any active else 0   (softmax mass surviving the mask)
// Pass C mapping: 8 waves x 32 lanes; lane owns a float4 of features
// (d = 4*dq..4*dq+3), wave g owns j = g mod 8. All per-j scalars are LDS
// broadcasts; zero-weight skip is wave-uniform. No atomics -> deterministic.
// ---------------------------------------------------------------------------
__global__ void __launch_bounds__(256)
attn_accum_kernel(const float* __restrict__ s1, const float* __restrict__ s2,
                  const int*   __restrict__ adj,
                  const float* __restrict__ pd, const float* __restrict__ ut,
                  const float* __restrict__ base, const float* __restrict__ W1,
                  float* __restrict__ acc, float* __restrict__ rowsum)
{
    __shared__ float  l_p [N_NODES];
    __shared__ float  l_pd[N_NODES];
    __shared__ float  l_ut[N_NODES];
    __shared__ float  red[256];
    __shared__ float4 sred4[256];
    __shared__ float  sh_max, sh_inv, sh_any;

    const int i   = blockIdx.x;
    const int tid = threadIdx.x;
    const float s1i = s1[i];
    const size_t rowOff = (size_t)i * N_NODES;

    // ---- pass A: row max over active entries ----
    float mloc = -3.4e38f;
    for (int j = tid; j < N_NODES; j += 256) {
        if (adj[rowOff + j] > 0) {
            float e = s1i + s2[j];
            e = (e >= 0.0f) ? e : 0.2f * e;
            mloc = fmaxf(mloc, e);
        }
    }
    red[tid] = mloc; __syncthreads();
    for (int s = 128; s > 0; s >>= 1) {
        if (tid < s) red[tid] = fmaxf(red[tid], red[tid + s]);
        __syncthreads();
    }
    if (tid == 0) sh_max = red[0];
    __syncthreads();
    const float rmax = sh_max;

    // ---- pass B: weights p[j], denom; stash per-j scalars in LDS ----
    float dloc = 0.0f;
    for (int j = tid; j < N_NODES; j += 256) {
        float p = 0.0f, pdv = 0.0f, utv = 0.0f;
        if (adj[rowOff + j] > 0) {
            float e = s1i + s2[j];
            e = (e >= 0.0f) ? e : 0.2f * e;
            p   = expf(e - rmax);
            pdv = pd[rowOff + j];
            utv = ut[rowOff + j];
        }
        l_p[j] = p; l_pd[j] = pdv; l_ut[j] = utv;
        dloc += p;
    }
    red[tid] = dloc; __syncthreads();
    for (int s = 128; s > 0; s >>= 1) {
        if (tid < s) red[tid] += red[tid + s];
        __syncthreads();
    }
    if (tid == 0) {
        const float denom = red[0];
        sh_any = (denom > 0.0f) ? 1.0f : 0.0f;
        sh_inv = (denom > 0.0f) ? (1.0f / denom) : 0.0f;
    }
    __syncthreads();
    const float inv = sh_inv;

    // ---- pass C: weighted relu accumulation over j (float4 per lane) ----
    const int dq = tid & 31;        // feature quad: d = 4*dq .. 4*dq+3
    const int g  = tid >> 5;        // j-subset; uniform across each wave
    const int d0 = dq * 4;

    float4 wpd, wut;
    wpd.x = W1[(size_t)(d0 + 0) * W1_LD + DIM];
    wpd.y = W1[(size_t)(d0 + 1) * W1_LD + DIM];
    wpd.z = W1[(size_t)(d0 + 2) * W1_LD + DIM];
    wpd.w = W1[(size_t)(d0 + 3) * W1_LD + DIM];
    wut.x = W1[(size_t)(d0 + 0) * W1_LD + DIM + 1];
    wut.y = W1[(size_t)(d0 + 1) * W1_LD + DIM + 1];
    wut.z = W1[(size_t)(d0 + 2) * W1_LD + DIM + 1];
    wut.w = W1[(size_t)(d0 + 3) * W1_LD + DIM + 1];

    float4 accv = make_float4(0.0f, 0.0f, 0.0f, 0.0f);
    for (int j = g; j < N_NODES; j += 8) {
        const float pj = l_p[j];    // LDS broadcast; branch uniform per wave
        if (pj != 0.0f) {
            const float pdj = l_pd[j];
            const float utj = l_ut[j];
            const float4 bv = *(const float4*)(base + (size_t)j * DIM + d0);
            float vx = fmaf(utj, wut.x, fmaf(pdj, wpd.x, bv.x));
            float vy = fmaf(utj, wut.y, fmaf(pdj, wpd.y, bv.y));
            float vz = fmaf(utj, wut.z, fmaf(pdj, wpd.z, bv.z));
            float vw = fmaf(utj, wut.w, fmaf(pdj, wpd.w, bv.w));
            accv.x = fmaf(pj, fmaxf(vx, 0.0f), accv.x);
            accv.y = fmaf(pj, fmaxf(vy, 0.0f), accv.y);
            accv.z = fmaf(pj, fmaxf(vz, 0.0f), accv.z);
            accv.w = fmaf(pj, fmaxf(vw, 0.0f), accv.w);
        }
    }
    sred4[tid] = accv;
    __syncthreads();

    if (g == 0) {
        float4 tot = sred4[dq];     // group 0
#pragma unroll
        for (int gg = 1; gg < 8; ++gg) {
            const float4 o = sred4[gg * 32 + dq];
            tot.x += o.x; tot.y += o.y; tot.z += o.z; tot.w += o.w;
        }
        float4 res = make_float4(tot.x * inv, tot.y * inv, tot.z * inv, tot.w * inv);
        *(float4*)(acc + (size_t)i * DIM + d0) = res;
        if (tid == 0) rowsum[i] = sh_any;
    }
}

// ---------------------------------------------------------------------------
extern "C" void kernel_launch(void* const* d_in, const int* in_sizes, int n_in,
                              void* d_out, int out_size, void* d_ws, size_t ws_size,
                              hipStream_t stream)
{
    const float* h   = (const float*)d_in[0];
    const int*   adj = (const int*)  d_in[1];
    const float* pd  = (const float*)d_in[2];
    const float* ut  = (const float*)d_in[3];
    const float* W   = (const float*)d_in[4];
    const float* a   = (const float*)d_in[5];
    const float* W1  = (const float*)d_in[6];
    const float* b1  = (const float*)d_in[7];
    const float* W2  = (const float*)d_in[8];
    const float* b2  = (const float*)d_in[9];
    float* out = (float*)d_out;

    float* ws   = (float*)d_ws;
    float* Wh   = ws;               // 1024*128
    float* base = ws + 131072;      // 1024*128
    float* acc  = ws + 262144;      // 1024*128
    float* s1   = ws + 393216;      // 1024
    float* s2   = ws + 394240;      // 1024
    float* rsum = ws + 395264;      // 1024   (total ~1.55 MB)

    const int tilesN = DIM / 16;                         // 8
    const int blocks = (N_NODES / 16) * tilesN / 8;      // 64 blocks, 1 wave/tile

    // 1. Wh = h @ W^T
    wmma_gemm_nt<<<blocks, 256, 0, stream>>>(h, DIM, W, DIM, Wh, DIM,
                                             nullptr, nullptr, DIM, tilesN);
    // 2. s1/s2 = Wh @ a[:D], Wh @ a[D:]
    s1s2_kernel<<<N_NODES, 256, 0, stream>>>(Wh, a, s1, s2);
    // 3. base = Wh @ W1[:, :D]^T + b1   (W1 row stride D+2)
    wmma_gemm_nt<<<blocks, 256, 0, stream>>>(Wh, DIM, W1, W1_LD, base, DIM,
                                             b1, nullptr, DIM, tilesN);
    // 4. masked softmax + weighted relu accumulation -> acc, rsum
    attn_accum_kernel<<<N_NODES, 256, 0, stream>>>(s1, s2, adj, pd, ut, base,
                                                   W1, acc, rsum);
    // 5. out = acc @ W2^T + rsum * b2
    wmma_gemm_nt<<<blocks, 256, 0, stream>>>(acc, DIM, W2, DIM, out, DIM,
                                             b2, rsum, DIM, tilesN);
}